// GraphDegreeConv_32847909880435
// MI455X (gfx1250) — compile-verified
//
#include <hip/hip_runtime.h>
#include <hip/hip_bf16.h>

typedef __attribute__((ext_vector_type(2))) float v2f;
typedef __attribute__((ext_vector_type(8))) float v8f;

#define NNODES   600000
#define NBUCKET  100000
#define BLK_ROWS 32
#define KSTRIDE  292      // 288 (=128 self + 160 neighbor) padded to avoid LDS bank conflicts
#define BN_EPS   1e-5f

// ---------------------------------------------------------------------------
// Fused kernel: builds combined A = [node_repr[row] | sum-gather(node,edge)]
// in LDS, then computes a = A @ [W_self ; W_d] + bias with fp32 WMMA.
// Grid: 6 buckets * 3125 blocks, 256 threads (8 waves) per block.
// Each block: 32 rows x 128 cols of output. Wave w: cols [16w,16w+16), 2 row tiles.
// ---------------------------------------------------------------------------
__global__ __launch_bounds__(256)
void gdc_fused_gemm(const float* __restrict__ node_repr,
                    const float* __restrict__ edge_repr,
                    const float* __restrict__ Wself,
                    const float* __restrict__ bias,
                    const float* __restrict__ Wd1, const int* __restrict__ ni1, const int* __restrict__ ei1,
                    const float* __restrict__ Wd2, const int* __restrict__ ni2, const int* __restrict__ ei2,
                    const float* __restrict__ Wd3, const int* __restrict__ ni3, const int* __restrict__ ei3,
                    const float* __restrict__ Wd4, const int* __restrict__ ni4, const int* __restrict__ ei4,
                    const float* __restrict__ Wd5, const int* __restrict__ ni5, const int* __restrict__ ei5,
                    float* __restrict__ out)
{
    __shared__ float lds[BLK_ROWS * KSTRIDE];   // 32*292*4 = 37376 bytes

    const int blocksPerBucket = NBUCKET / BLK_ROWS;           // 3125
    const int bucket = blockIdx.x / blocksPerBucket;          // 0..5 == degree
    const int lblk   = blockIdx.x % blocksPerBucket;
    const int row0   = bucket * NBUCKET + lblk * BLK_ROWS;    // global output row base
    const int i0     = lblk * BLK_ROWS;                       // bucket-local row base
    const int deg    = bucket;

    const float* Wd = Wself;                 // never dereferenced when deg==0
    const int* nidx = nullptr;
    const int* eidx = nullptr;
    switch (bucket) {
      case 1: Wd = Wd1; nidx = ni1; eidx = ei1; break;
      case 2: Wd = Wd2; nidx = ni2; eidx = ei2; break;
      case 3: Wd = Wd3; nidx = ni3; eidx = ei3; break;
      case 4: Wd = Wd4; nidx = ni4; eidx = ei4; break;
      case 5: Wd = Wd5; nidx = ni5; eidx = ei5; break;
      default: break;
    }

    const int t = threadIdx.x;

    // ---- stage self part: lds[r][0..127] = node_repr[row0+r][:]  (32 rows * 32 float4)
    #pragma unroll
    for (int it = 0; it < 4; ++it) {
        int slot = t + 256 * it;                 // 0..1023
        int r    = slot >> 5;
        int c4   = (slot & 31) << 2;
        float4 v = *(const float4*)&node_repr[(size_t)(row0 + r) * 128 + c4];
        *(float4*)&lds[r * KSTRIDE + c4] = v;
    }

    // ---- gather-sum neighbor part into lds[r][128 + c], c in [0,160)
    if (deg > 0) {
        int r = t >> 3;        // 8 threads per row
        int p = t & 7;         // each part covers 20 columns (5 float4)
        int i = i0 + r;
        int ni[5], ei[5];
        for (int j = 0; j < deg; ++j) {
            ni[j] = nidx[i * deg + j];
            ei[j] = eidx[i * deg + j];
        }
        #pragma unroll
        for (int s = 0; s < 5; ++s) {
            int col = p * 20 + s * 4;            // 0..156, float4 never crosses 128
            float4 acc = make_float4(0.f, 0.f, 0.f, 0.f);
            if (col < 128) {
                for (int j = 0; j < deg; ++j) {
                    float4 v = *(const float4*)&node_repr[(size_t)ni[j] * 128 + col];
                    acc.x += v.x; acc.y += v.y; acc.z += v.z; acc.w += v.w;
                }
            } else {
                int ec = col - 128;
                for (int j = 0; j < deg; ++j) {
                    float4 v = *(const float4*)&edge_repr[(size_t)ei[j] * 32 + ec];
                    acc.x += v.x; acc.y += v.y; acc.z += v.z; acc.w += v.w;
                }
            }
            *(float4*)&lds[r * KSTRIDE + 128 + col] = acc;
        }
    }
    __syncthreads();

    // ---- WMMA fp32 GEMM: D[32x16 per wave] = A[32xK] * W[Kx16] ------------
    // Two uniform phases (self K=128, neighbor K=160) so B addresses are
    // SGPR base + imm + loop-invariant lane offset: no branches in the loop.
    const int lane = t & 31;
    const int wv   = t >> 5;                 // wave id 0..7
    const int n0   = wv * 16;                // column tile base
    const int m    = lane & 15;              // row within 16-row tile / col within strip
    const int qoff = (lane >> 4) << 1;       // 0 or 2 (A/B K-pair select per ISA layout)
    const int n    = n0 + m;                 // output column for B-frag and store

    const int aoff = m * KSTRIDE + qoff;     // loop-invariant A LDS offset (floats)
    const int boff = qoff * 128 + n;         // loop-invariant B offset (floats)

    v8f acc0 = {0.f,0.f,0.f,0.f,0.f,0.f,0.f,0.f};
    v8f acc1 = {0.f,0.f,0.f,0.f,0.f,0.f,0.f,0.f};

    // Phase 1: self contribution, K = 0..127 over Wself
    {
        const float* wb = Wself + boff;
        #pragma unroll 8
        for (int k0 = 0; k0 < 128; k0 += 4) {
            v2f a0 = *(const v2f*)&lds[aoff + k0];
            v2f a1 = *(const v2f*)&lds[aoff + 16 * KSTRIDE + k0];
            v2f b;
            b.x = wb[k0 * 128];
            b.y = wb[k0 * 128 + 128];
            acc0 = __builtin_amdgcn_wmma_f32_16x16x4_f32(false, a0, false, b,
                                                         (short)0, acc0, false, false);
            acc1 = __builtin_amdgcn_wmma_f32_16x16x4_f32(false, a1, false, b,
                                                         (short)0, acc1, false, false);
        }
    }

    // Phase 2: neighbor contribution, K = 0..159 over Wd (block-uniform guard)
    if (deg > 0) {
        const float* wb = Wd + boff;
        #pragma unroll 8
        for (int k0 = 0; k0 < 160; k0 += 4) {
            v2f a0 = *(const v2f*)&lds[aoff + 128 + k0];
            v2f a1 = *(const v2f*)&lds[aoff + 16 * KSTRIDE + 128 + k0];
            v2f b;
            b.x = wb[k0 * 128];
            b.y = wb[k0 * 128 + 128];
            acc0 = __builtin_amdgcn_wmma_f32_16x16x4_f32(false, a0, false, b,
                                                         (short)0, acc0, false, false);
            acc1 = __builtin_amdgcn_wmma_f32_16x16x4_f32(false, a1, false, b,
                                                         (short)0, acc1, false, false);
        }
    }

    // ---- store a = gemm + bias (C layout: VGPR v -> rows v and v+8) -------
    const float bn = bias[n];
    const int mhi = (lane >> 4) << 3;        // 0 or 8
    #pragma unroll
    for (int v = 0; v < 8; ++v) {
        int M = v + mhi;
        out[(size_t)(row0 + M) * 128 + n]      = acc0[v] + bn;
        out[(size_t)(row0 + 16 + M) * 128 + n] = acc1[v] + bn;
    }
}

// ---------------------------------------------------------------------------
// Column sum / sum-of-squares reduction for BatchNorm (ws[0:128]=sum, ws[128:256]=sumsq)
// ---------------------------------------------------------------------------
__global__ __launch_bounds__(256)
void gdc_colsum(const float* __restrict__ a, float* __restrict__ ws)
{
    __shared__ float s1[256], s2[256];
    const int c    = threadIdx.x & 127;
    const int half = threadIdx.x >> 7;
    int chunk = (NNODES + gridDim.x - 1) / gridDim.x;
    int r0 = blockIdx.x * chunk;
    int r1 = r0 + chunk; if (r1 > NNODES) r1 = NNODES;
    float s = 0.f, q = 0.f;
    for (int r = r0 + half; r < r1; r += 2) {
        float v = a[(size_t)r * 128 + c];
        s += v; q += v * v;
    }
    s1[threadIdx.x] = s;
    s2[threadIdx.x] = q;
    __syncthreads();
    if (half == 0) {
        atomicAdd(&ws[c],       s + s1[c + 128]);
        atomicAdd(&ws[128 + c], q + s2[c + 128]);
    }
}

// ---------------------------------------------------------------------------
// In-place BatchNorm (affine=False, biased var) + ReLU
// ---------------------------------------------------------------------------
__global__ __launch_bounds__(256)
void gdc_bn_relu(float* __restrict__ a, const float* __restrict__ ws)
{
    const float invN = 1.0f / (float)NNODES;
    const size_t total4 = (size_t)NNODES * 32;     // elements / 4
    size_t idx = (size_t)blockIdx.x * blockDim.x + threadIdx.x;
    const size_t stride = (size_t)gridDim.x * blockDim.x;
    for (; idx < total4; idx += stride) {
        int c = (int)((idx & 31) << 2);            // (idx*4) % 128
        float4 v = *(float4*)&a[idx * 4];
        float m, var, sc;
        m = ws[c+0]*invN; var = ws[128+c+0]*invN - m*m; sc = rsqrtf(var + BN_EPS);
        v.x = fmaxf(0.f, (v.x - m) * sc);
        m = ws[c+1]*invN; var = ws[128+c+1]*invN - m*m; sc = rsqrtf(var + BN_EPS);
        v.y = fmaxf(0.f, (v.y - m) * sc);
        m = ws[c+2]*invN; var = ws[128+c+2]*invN - m*m; sc = rsqrtf(var + BN_EPS);
        v.z = fmaxf(0.f, (v.z - m) * sc);
        m = ws[c+3]*invN; var = ws[128+c+3]*invN - m*m; sc = rsqrtf(var + BN_EPS);
        v.w = fmaxf(0.f, (v.w - m) * sc);
        *(float4*)&a[idx * 4] = v;
    }
}

__global__ void gdc_zero(float* __restrict__ ws) { ws[threadIdx.x] = 0.f; }

// ---------------------------------------------------------------------------
extern "C" void kernel_launch(void* const* d_in, const int* in_sizes, int n_in,
                              void* d_out, int out_size, void* d_ws, size_t ws_size,
                              hipStream_t stream)
{
    const float* node  = (const float*)d_in[0];
    const float* edge  = (const float*)d_in[1];
    const float* Wself = (const float*)d_in[2];
    const float* bias  = (const float*)d_in[3];
    const float* Wd[5]; const int* ni[5]; const int* ei[5];
    for (int k = 0; k < 5; ++k) {
        Wd[k] = (const float*)d_in[4 + 3*k];
        ni[k] = (const int*)  d_in[5 + 3*k];
        ei[k] = (const int*)  d_in[6 + 3*k];
    }
    float* out = (float*)d_out;
    float* ws  = (float*)d_ws;

    gdc_zero<<<1, 256, 0, stream>>>(ws);

    const int grid = 6 * (NBUCKET / BLK_ROWS);    // 18750 blocks
    gdc_fused_gemm<<<grid, 256, 0, stream>>>(node, edge, Wself, bias,
        Wd[0], ni[0], ei[0], Wd[1], ni[1], ei[1], Wd[2], ni[2], ei[2],
        Wd[3], ni[3], ei[3], Wd[4], ni[4], ei[4], out);

    gdc_colsum<<<1024, 256, 0, stream>>>(out, ws);
    gdc_bn_relu<<<4096, 256, 0, stream>>>(out, ws);
}